// MPNN_49409303773430
// MI455X (gfx1250) — compile-verified
//
#include <hip/hip_runtime.h>
#include <math.h>

#define N_NODES 20000
#define N_EDGES 640000
#define HID     128
#define HOPS    3
#define LEAKY   0.2f

typedef float v2f __attribute__((ext_vector_type(2)));
typedef float v8f __attribute__((ext_vector_type(8)));

// Monotone order-preserving float<->uint mapping for atomicMax on floats
// (incl. negatives): positive -> set MSB, negative -> bitwise NOT.
__device__ __forceinline__ unsigned flipF(float f) {
    unsigned u = __float_as_uint(f);
    return (u & 0x80000000u) ? ~u : (u | 0x80000000u);
}
__device__ __forceinline__ float unflipF(unsigned u) {
    u = (u & 0x80000000u) ? (u ^ 0x80000000u) : ~u;
    return __uint_as_float(u);
}

// ---------------------------------------------------------------------------
// LDS staging in row-paired layout:
//   sW[p*256 + col*2 + r] = W[kc + 2p + r][col],  p in [0,16), r in {0,1}
// so the (K,K+1) B-operand pair for any lane is one aligned ds_load_b64.
// ---------------------------------------------------------------------------
__device__ __forceinline__ void stage_W_paired(float* sW, const float* __restrict__ W,
                                               int kc, int tid)
{
#pragma unroll
    for (int t = 0; t < 16; ++t) {
        const int i = tid + 128 * t;        // 0..2047
        const int p = i >> 7;               // row-pair index 0..15
        const int c = i & 127;              // column
        float2 w;
        w.x = W[(kc + 2 * p) * HID + c];
        w.y = W[(kc + 2 * p + 1) * HID + c];
        ((float2*)sW)[p * 128 + c] = w;     // contiguous 8B store
    }
}

// ---------------------------------------------------------------------------
// Y[M,128] = X[M,128] @ W[128,128] + bias        (f32 WMMA 16x16x4)
// Block = 128 threads = 4 waves; each wave owns a 16-row x 128-col strip.
// W staged through LDS in 32-row chunks shared by all 4 waves.
// ---------------------------------------------------------------------------
__global__ __launch_bounds__(128) void linear_kernel(
    const float* __restrict__ X, const float* __restrict__ W,
    const float* __restrict__ bias, float* __restrict__ Y, int M)
{
    __shared__ float sW[32 * 128];
    const int lane = threadIdx.x & 31;
    const int wave = threadIdx.x >> 5;
    const int half = lane >> 4;     // 0: K+0/K+1 , 1: K+2/K+3 (A); rows +0..7 vs +8..15 (D)
    const int l    = lane & 15;
    const int tile = blockIdx.x * 4 + wave;
    const int numTiles = M >> 4;
    const bool valid = tile < numTiles;
    const int r0 = tile << 4;

    const v8f vzero = {0.f,0.f,0.f,0.f,0.f,0.f,0.f,0.f};
    v8f acc[8];
#pragma unroll
    for (int c = 0; c < 8; ++c) acc[c] = vzero;

    for (int kc = 0; kc < HID; kc += 32) {
        __syncthreads();
        stage_W_paired(sW, W, kc, threadIdx.x);
        __syncthreads();
        if (valid) {
#pragma unroll
            for (int ks = 0; ks < 32; ks += 4) {
                const int k = kc + ks;
                // A 16x4 f32: lanes 0-15: K=k,k+1 ; lanes 16-31: K=k+2,k+3
                const float* xp = X + (size_t)(r0 + l) * HID + k + 2 * half;
                v2f a; a.x = xp[0]; a.y = xp[1];
                const float* wrow = sW + (ks / 2 + half) * 256;   // pair (k+2*half, +1)
#pragma unroll
                for (int c = 0; c < 8; ++c) {
                    v2f b = *(const v2f*)(wrow + (c * 16 + l) * 2);  // ds_load_b64
                    acc[c] = __builtin_amdgcn_wmma_f32_16x16x4_f32(
                        false, a, false, b, (short)0, acc[c], false, false);
                }
            }
        }
    }
    if (valid) {
#pragma unroll
        for (int c = 0; c < 8; ++c) {
            const int col = c * 16 + l;
            const float bb = bias[col];
#pragma unroll
            for (int j = 0; j < 8; ++j) {
                const int row = r0 + j + 8 * half;
                Y[(size_t)row * HID + col] = acc[c][j] + bb;
            }
        }
    }
}

// ---------------------------------------------------------------------------
// Edge GEMM + fused value assembly + attention score + segment-max.
//  v[e]   = edge_feat[e] @ We + be + vn[senders[e]]          (stored)
//  s[e]   = attn . leaky_relu(qn[receivers[e]] + v[e])       (stored)
//  mkey[recv] = atomicMax(flip(s))
// ---------------------------------------------------------------------------
__global__ __launch_bounds__(128) void edge_kernel(
    const float* __restrict__ EF, const float* __restrict__ W,
    const float* __restrict__ bias, const float* __restrict__ vn,
    const float* __restrict__ qn, const float* __restrict__ attn,
    const int* __restrict__ senders, const int* __restrict__ recvs,
    float* __restrict__ V, float* __restrict__ score,
    unsigned* __restrict__ mkey)
{
    __shared__ float sW[32 * 128];
    const int lane = threadIdx.x & 31;
    const int wave = threadIdx.x >> 5;
    const int half = lane >> 4;
    const int l    = lane & 15;
    const int tile = blockIdx.x * 4 + wave;
    const bool valid = tile < (N_EDGES >> 4);
    const int r0 = tile << 4;

    const v8f vzero = {0.f,0.f,0.f,0.f,0.f,0.f,0.f,0.f};
    v8f acc[8];
#pragma unroll
    for (int c = 0; c < 8; ++c) acc[c] = vzero;

    for (int kc = 0; kc < HID; kc += 32) {
        __syncthreads();
        stage_W_paired(sW, W, kc, threadIdx.x);
        __syncthreads();
        if (valid) {
#pragma unroll
            for (int ks = 0; ks < 32; ks += 4) {
                const int k = kc + ks;
                const float* xp = EF + (size_t)(r0 + l) * HID + k + 2 * half;
                v2f a; a.x = xp[0]; a.y = xp[1];
                const float* wrow = sW + (ks / 2 + half) * 256;
#pragma unroll
                for (int c = 0; c < 8; ++c) {
                    v2f b = *(const v2f*)(wrow + (c * 16 + l) * 2);  // ds_load_b64
                    acc[c] = __builtin_amdgcn_wmma_f32_16x16x4_f32(
                        false, a, false, b, (short)0, acc[c], false, false);
                }
            }
        }
    }
    if (valid) {
        float av[8], bv[8];
#pragma unroll
        for (int c = 0; c < 8; ++c) { av[c] = attn[c * 16 + l]; bv[c] = bias[c * 16 + l]; }
#pragma unroll
        for (int j = 0; j < 8; ++j) {
            const int row = r0 + j + 8 * half;      // edge index
            const int snd = senders[row];
            const int rcv = recvs[row];
            const float* vnp = vn + (size_t)snd * HID;
            const float* qnp = qn + (size_t)rcv * HID;
            float partial = 0.f;
#pragma unroll
            for (int c = 0; c < 8; ++c) {
                const int col = c * 16 + l;
                const float val = acc[c][j] + bv[c] + vnp[col];
                V[(size_t)row * HID + col] = val;
                float f = val + qnp[col];
                f = (f > 0.f) ? f : LEAKY * f;
                partial += av[c] * f;
            }
            // reduce across the 16 lanes of this half (rows differ per half)
            partial += __shfl_xor(partial, 1);
            partial += __shfl_xor(partial, 2);
            partial += __shfl_xor(partial, 4);
            partial += __shfl_xor(partial, 8);
            if (l == 0) {
                score[row] = partial;
                atomicMax(mkey + rcv, flipF(partial));
            }
        }
    }
}

// score[e] <- exp(score[e] - max[recv]); denom[recv] += score[e]
__global__ __launch_bounds__(256) void expsum_kernel(
    float* __restrict__ score, const int* __restrict__ recvs,
    const unsigned* __restrict__ mkey, float* __restrict__ denom)
{
    int e = blockIdx.x * 256 + threadIdx.x;
    if (e < N_EDGES) {
        int r = recvs[e];
        float ex = expf(score[e] - unflipF(mkey[r]));
        score[e] = ex;
        unsafeAtomicAdd(denom + r, ex);
    }
}

// hacc[recv] += (score[e]/denom[recv]) * V[e]   — one wave per edge
__global__ __launch_bounds__(256) void scatter_kernel(
    const float* __restrict__ V, const float* __restrict__ score,
    const float* __restrict__ denom, const int* __restrict__ recvs,
    float* __restrict__ hacc)
{
    const int lane = threadIdx.x & 31;
    const int e = blockIdx.x * 8 + (threadIdx.x >> 5);
    if (e >= N_EDGES) return;
    const int r = recvs[e];
    const float alpha = score[e] / denom[r];
    const float4 x = ((const float4*)(V + (size_t)e * HID))[lane];
    float* hp = hacc + (size_t)r * HID + lane * 4;
    unsafeAtomicAdd(hp + 0, alpha * x.x);
    unsafeAtomicAdd(hp + 1, alpha * x.y);
    unsafeAtomicAdd(hp + 2, alpha * x.z);
    unsafeAtomicAdd(hp + 3, alpha * x.w);
}

__global__ __launch_bounds__(256) void init_kernel(
    float* __restrict__ hacc, unsigned* __restrict__ mkey, float* __restrict__ denom)
{
    int i = blockIdx.x * 256 + threadIdx.x;
    if (i < N_NODES * HID) hacc[i] = 0.f;
    if (i < N_NODES) { mkey[i] = 0x007FFFFFu; /* flip(-inf) */ denom[i] = 0.f; }
}

__global__ __launch_bounds__(256) void relu_kernel(
    const float* __restrict__ hacc, float* __restrict__ out)
{
    int i = blockIdx.x * 256 + threadIdx.x;
    if (i < N_NODES * HID) out[i] = fmaxf(hacc[i], 0.f);
}

extern "C" void kernel_launch(void* const* d_in, const int* in_sizes, int n_in,
                              void* d_out, int out_size, void* d_ws, size_t ws_size,
                              hipStream_t stream)
{
    const float* node_feat = (const float*)d_in[0];
    const float* edge_feat = (const float*)d_in[1];
    const int*   senders   = (const int*)d_in[2];
    const int*   recvs     = (const int*)d_in[3];
    const float* Wq = (const float*)d_in[4];
    const float* bq = (const float*)d_in[5];
    const float* Wn = (const float*)d_in[6];
    const float* bn = (const float*)d_in[7];
    const float* We = (const float*)d_in[8];
    const float* be = (const float*)d_in[9];
    const float* attn = (const float*)d_in[10];

    float* ws = (float*)d_ws;
    size_t off = 0;
    float*    qn    = ws + off; off += (size_t)N_NODES * HID;
    float*    vn    = ws + off; off += (size_t)N_NODES * HID;
    float*    V     = ws + off; off += (size_t)N_EDGES * HID;
    float*    score = ws + off; off += (size_t)N_EDGES;
    unsigned* mkey  = (unsigned*)(ws + off); off += (size_t)N_NODES;
    float*    denom = ws + off; off += (size_t)N_NODES;
    float*    hacc  = ws + off; off += (size_t)N_NODES * HID;
    float*    hA    = ws + off; off += (size_t)N_NODES * HID;
    float*    hB    = ws + off; off += (size_t)N_NODES * HID;

    const int nodeTiles = N_NODES / 16;                 // 1250
    const int nodeBlocks = (nodeTiles + 3) / 4;         // 313 (tail-guarded)
    const int edgeBlocks = N_EDGES / 64;                // 10000
    const int nhElems = N_NODES * HID;

    for (int i = 0; i < HOPS; ++i) {
        const float* hin  = (i == 0) ? node_feat : ((i == 1) ? hA : hB);
        float*       hout = (i == 0) ? hA : ((i == 1) ? hB : (float*)d_out);
        const float* Wq_i = Wq + (size_t)i * HID * HID;
        const float* Wn_i = Wn + (size_t)i * HID * HID;
        const float* We_i = We + (size_t)i * HID * HID;
        const float* bq_i = bq + (size_t)i * HID;
        const float* bn_i = bn + (size_t)i * HID;
        const float* be_i = be + (size_t)i * HID;
        const float* at_i = attn + (size_t)i * HID;

        init_kernel<<<(nhElems + 255) / 256, 256, 0, stream>>>(hacc, mkey, denom);
        linear_kernel<<<nodeBlocks, 128, 0, stream>>>(hin, Wq_i, bq_i, qn, N_NODES);
        linear_kernel<<<nodeBlocks, 128, 0, stream>>>(hin, Wn_i, bn_i, vn, N_NODES);
        edge_kernel<<<edgeBlocks, 128, 0, stream>>>(edge_feat, We_i, be_i, vn, qn, at_i,
                                                    senders, recvs, V, score, mkey);
        expsum_kernel<<<(N_EDGES + 255) / 256, 256, 0, stream>>>(score, recvs, mkey, denom);
        scatter_kernel<<<N_EDGES / 8, 256, 0, stream>>>(V, score, denom, recvs, hacc);
        relu_kernel<<<(nhElems + 255) / 256, 256, 0, stream>>>(hacc, hout);
    }
}